// UtilityInfrastructureBalancer_11450382811460
// MI455X (gfx1250) — compile-verified
//
#include <hip/hip_runtime.h>
#include <hip/hip_bf16.h>

// N = 512. Softmin Floyd-Warshall via exp-domain semiring transform:
//   e = 2^112 * exp(-w/gamma);  softmin-FW step == e += e[:,k] e[k,:]
// Blocked like right-looking LU: LDS panel sweep (exact sequential recurrence
// on the panels, recording pre-step snapshots) + trailing GEMM E += C~ R~
// executed with V_WMMA_F32_16X16X4_F32. Each GEMM wave owns a 2x2 group of
// 16x16 tiles (16 WMMAs, A/B fragment reuse across the group).

#define NN 512
#define NB 16            // K-block (panel) width
#define NBLK (NN / NB)   // 32 sequential blocks
#define FSCALE_LOG2 112.0f
#define FSCALE_INV  0x1p-112f
#define LOG2E_X10   14.4269504088896f   // 10/ln2  (gamma = 0.1)
#define GAMMA_LN2   0.0693147180560f    // gamma*ln2

typedef float v2f __attribute__((ext_vector_type(2)));
typedef float v8f __attribute__((ext_vector_type(8)));

__device__ __forceinline__ void blockReduceAdd256(float v, float* out, float* sdata) {
    const int t = threadIdx.x;
    sdata[t] = v;
    __syncthreads();
    for (int s = 128; s > 0; s >>= 1) {
        if (t < s) sdata[t] += sdata[t + s];
        __syncthreads();
    }
    if (t == 0) out[blockIdx.x] = sdata[0];
}

// ---------------------------------------------------------------------------
// Kernel 1: build f = 2^112 * exp(-10*w), w = (dist/(soft+1e-4))*(1-eye),
// and reduce the sp-independent loss terms (cost + entropy + mask).
// ---------------------------------------------------------------------------
__global__ __launch_bounds__(256) void setup_kernel(
    const float* __restrict__ soft, const float* __restrict__ orig,
    const float* __restrict__ dist, const int* __restrict__ epoch,
    float* __restrict__ E, float* __restrict__ partial1) {
    __shared__ float sdata[256];
    const int idx = blockIdx.x * 256 + threadIdx.x;
    const int i = idx >> 9, j = idx & (NN - 1);
    const float s = soft[idx], d = dist[idx], o = orig[idx];
    const int ep = epoch[0];
    // searchsorted([0,100], ep, 'right') -> levels [0.1, 0.5, 1.0]; same for mask
    const float sc = (ep >= 100) ? 1.0f : ((ep >= 0) ? 0.5f : 0.1f);
    const float w = (i == j) ? 0.0f : d / (s + 1e-4f);
    E[idx] = exp2f(FSCALE_LOG2 - LOG2E_X10 * w);   // no intermediate underflow
    const float eye = (i == j) ? 1.0f : 0.0f;
    const float t = s * (eye - s);
    const float contrib = s * d + sc * t * t + sc * s * (1.0f - o);
    blockReduceAdd256(contrib, partial1, sdata);
}

// ---------------------------------------------------------------------------
// Kernel 2: panel sweep for K-block [base, base+16). One workgroup, 1024
// threads, panels resident in LDS (2 x 32KB). For each k in block: snapshot
// pre-update column/row (the GEMM multiplicands), then apply the rank-1
// update to both panels (diag block kept coherent in both copies).
// ---------------------------------------------------------------------------
__global__ __launch_bounds__(1024) void panel_kernel(
    float* __restrict__ E, float* __restrict__ Ct, float* __restrict__ Rt,
    int base) {
    __shared__ float colP[NN * NB];   // [512][16]  E[:, base..base+15]
    __shared__ float rowP[NB * NN];   // [16][512]  E[base..base+15, :]
    const int tid = threadIdx.x;
    const int tx = tid & (NB - 1);    // 0..15  panel column
    const int ty = tid >> 4;          // 0..63  row group (colP)
    const int r  = tid >> 6;          // 0..15  panel row  (rowP)
    const int jx = tid & 63;          // 0..63  col group  (rowP)

    #pragma unroll
    for (int m = 0; m < 8; ++m) {
        const int i = ty + 64 * m;
        colP[i * NB + tx] = E[i * NN + base + tx];
        const int j = jx + 64 * m;
        rowP[r * NN + j]  = E[(base + r) * NN + j];
    }
    __syncthreads();

    for (int k = 0; k < NB; ++k) {
        float a[8], h[8];
        #pragma unroll
        for (int m = 0; m < 8; ++m) {
            a[m] = colP[(ty + 64 * m) * NB + k];   // pre-update col k
            h[m] = rowP[k * NN + jx + 64 * m];     // pre-update row k
        }
        const float dmul = rowP[k * NN + base + tx] * FSCALE_INV;
        const float gmul = colP[(base + r) * NB + k] * FSCALE_INV;
        if (tx == k) {
            #pragma unroll
            for (int m = 0; m < 8; ++m) Ct[k * NN + ty + 64 * m] = a[m];
        }
        if (r == k) {
            #pragma unroll
            for (int m = 0; m < 8; ++m) Rt[k * NN + jx + 64 * m] = h[m];
        }
        __syncthreads();                 // all pre-values read before writes
        #pragma unroll
        for (int m = 0; m < 8; ++m) {
            colP[(ty + 64 * m) * NB + tx] += a[m] * dmul;
            rowP[r * NN + jx + 64 * m]    += gmul * h[m];
        }
        __syncthreads();
    }

    #pragma unroll
    for (int m = 0; m < 8; ++m) {
        const int i = ty + 64 * m;
        E[i * NN + base + tx] = colP[i * NB + tx];
        const int j = jx + 64 * m;
        E[(base + r) * NN + j] = rowP[r * NN + j];
    }
}

// ---------------------------------------------------------------------------
// Kernel 3: trailing update E += (C~ * 2^-112) x R~. Each wave owns a 2x2
// group of 16x16 tiles: A fragments reused across two column tiles, B
// fragments across two row tiles -> 16 WMMAs per wave. Tiles lying in the
// panel row/col were finalized in phase 1: computed here but never stored
// (store guards are wave-uniform; EXEC stays all-ones at every WMMA).
// ---------------------------------------------------------------------------
__global__ __launch_bounds__(256) void gemm_kernel(
    float* __restrict__ E, const float* __restrict__ Ct,
    const float* __restrict__ Rt, int base) {
    const int lane = threadIdx.x & 31;
    const int wave = threadIdx.x >> 5;          // 8 waves / block
    const int grp  = blockIdx.x * 8 + wave;     // 0..255 (16x16 groups)
    const int gi = grp >> 4, gj = grp & 15;
    const int r0 = gi << 1, r1 = r0 + 1;        // row tile indices (0..31)
    const int c0 = gj << 1, c1 = c0 + 1;        // col tile indices (0..31)
    const int pb = base >> 4;                   // panel tile index
    const int ib0 = r0 << 4, ib1 = r1 << 4;
    const int jb0 = c0 << 4, jb1 = c1 << 4;
    const int lrow   = lane & 15;
    const int khalf  = (lane >> 4) << 1;        // A/B: lanes<16 K{0,1}, else K{2,3}
    const int rowoff = (lane >> 4) << 3;        // C/D: lanes<16 M g, else M g+8

    v8f c00, c01, c10, c11;
    #pragma unroll
    for (int g = 0; g < 8; ++g) {
        c00[g] = E[(ib0 + rowoff + g) * NN + jb0 + lrow];
        c01[g] = E[(ib0 + rowoff + g) * NN + jb1 + lrow];
        c10[g] = E[(ib1 + rowoff + g) * NN + jb0 + lrow];
        c11[g] = E[(ib1 + rowoff + g) * NN + jb1 + lrow];
    }

    #pragma unroll
    for (int k0 = 0; k0 < NB; k0 += 4) {
        v2f a0, a1, b0, b1;
        a0[0] = Ct[(k0 + khalf + 0) * NN + ib0 + lrow] * FSCALE_INV;
        a0[1] = Ct[(k0 + khalf + 1) * NN + ib0 + lrow] * FSCALE_INV;
        a1[0] = Ct[(k0 + khalf + 0) * NN + ib1 + lrow] * FSCALE_INV;
        a1[1] = Ct[(k0 + khalf + 1) * NN + ib1 + lrow] * FSCALE_INV;
        b0[0] = Rt[(k0 + khalf + 0) * NN + jb0 + lrow];
        b0[1] = Rt[(k0 + khalf + 1) * NN + jb0 + lrow];
        b1[0] = Rt[(k0 + khalf + 0) * NN + jb1 + lrow];
        b1[1] = Rt[(k0 + khalf + 1) * NN + jb1 + lrow];
        c00 = __builtin_amdgcn_wmma_f32_16x16x4_f32(false, a0, false, b0, (short)0, c00, false, false);
        c01 = __builtin_amdgcn_wmma_f32_16x16x4_f32(false, a0, false, b1, (short)0, c01, false, false);
        c10 = __builtin_amdgcn_wmma_f32_16x16x4_f32(false, a1, false, b0, (short)0, c10, false, false);
        c11 = __builtin_amdgcn_wmma_f32_16x16x4_f32(false, a1, false, b1, (short)0, c11, false, false);
    }

    const bool sr0 = (r0 != pb), sr1 = (r1 != pb);
    const bool sc0 = (c0 != pb), sc1 = (c1 != pb);
    #pragma unroll
    for (int g = 0; g < 8; ++g) {
        if (sr0 && sc0) E[(ib0 + rowoff + g) * NN + jb0 + lrow] = c00[g];
        if (sr0 && sc1) E[(ib0 + rowoff + g) * NN + jb1 + lrow] = c01[g];
        if (sr1 && sc0) E[(ib1 + rowoff + g) * NN + jb0 + lrow] = c10[g];
        if (sr1 && sc1) E[(ib1 + rowoff + g) * NN + jb1 + lrow] = c11[g];
    }
}

// ---------------------------------------------------------------------------
// Kernel 4: recover sp = -gamma*ln2*(log2 f - 112), compute utility_gain,
// block-reduce. f==0 -> sp=+inf -> choice=0, elu+alpha=0 (no NaNs).
// ---------------------------------------------------------------------------
__global__ __launch_bounds__(256) void utility_kernel(
    const float* __restrict__ E, const float* __restrict__ dist,
    const float* __restrict__ flow, float* __restrict__ partial2) {
    __shared__ float sdata[256];
    const int idx = blockIdx.x * 256 + threadIdx.x;
    const int i = idx >> 9, j = idx & (NN - 1);
    const float f = E[idx];
    const float sp = -GAMMA_LN2 * (log2f(f) - FSCALE_LOG2);
    const float d = dist[idx];
    const float rail  = expf(-0.005f * sp);     // exp(UTIL_SCALE*sp*RAIL)
    const float basev = expf(-0.01f * d);
    const float choice = rail / (rail + basev);
    const float x = d - 0.5f * sp;
    float dsaved = (x > 0.0f) ? (x + 1.0f) : expf(x);  // elu(x,1)+1
    if (i == j) dsaved = 0.0f;
    blockReduceAdd256(flow[idx] * choice * dsaved, partial2, sdata);
}

// ---------------------------------------------------------------------------
// Kernel 5: deterministic final reduction of both partial arrays (fixed
// order, no float atomics -> reproducible across graph replays).
// ---------------------------------------------------------------------------
__global__ __launch_bounds__(256) void final_kernel(
    const float* __restrict__ p1, const float* __restrict__ p2,
    float* __restrict__ out) {
    __shared__ float sdata[256];
    const int t = threadIdx.x;
    float v = 0.0f;
    #pragma unroll
    for (int m = 0; m < 4; ++m) v += p1[t + 256 * m] + p2[t + 256 * m];
    sdata[t] = v;
    __syncthreads();
    for (int s = 128; s > 0; s >>= 1) {
        if (t < s) sdata[t] += sdata[t + s];
        __syncthreads();
    }
    if (t == 0) out[0] = sdata[0];
}

extern "C" void kernel_launch(void* const* d_in, const int* in_sizes, int n_in,
                              void* d_out, int out_size, void* d_ws, size_t ws_size,
                              hipStream_t stream) {
    const float* soft  = (const float*)d_in[0];
    const float* orig  = (const float*)d_in[1];
    const float* dist  = (const float*)d_in[2];
    const float* flow  = (const float*)d_in[3];
    const int*   epoch = (const int*)d_in[4];

    float* ws = (float*)d_ws;
    float* E  = ws;                 // 512*512 floats (1 MB)
    float* Ct = E  + NN * NN;       // snapshots C~ stored [k][i], 16*512
    float* Rt = Ct + NB * NN;       // snapshots R~ stored [k][j], 16*512
    float* p1 = Rt + NB * NN;       // 1024 partials (static losses)
    float* p2 = p1 + 1024;          // 1024 partials (utility)

    setup_kernel<<<1024, 256, 0, stream>>>(soft, orig, dist, epoch, E, p1);
    for (int bb = 0; bb < NBLK; ++bb) {
        panel_kernel<<<1, 1024, 0, stream>>>(E, Ct, Rt, bb * NB);
        gemm_kernel<<<32, 256, 0, stream>>>(E, Ct, Rt, bb * NB);
    }
    utility_kernel<<<1024, 256, 0, stream>>>(E, dist, flow, p2);
    final_kernel<<<1, 256, 0, stream>>>(p1, p2, (float*)d_out);
}